// CRF_SECLoss_vb_73315091742830
// MI455X (gfx1250) — compile-verified
//
#include <hip/hip_runtime.h>
#include <cstdint>

// CRF SECLoss for MI455X (gfx1250).
// scores: [128,32,32,32,32] f32 (512 MB), target: [128,32,1] i64, mask: [128,32] bool
// Memory-bound streaming scan: double-buffered async global->LDS (CDNA5 ASYNCcnt path),
// one workgroup per batch lane, 32 waves (wave32) = 1024 threads, LDS = 2*128KB + 4KB.

#define AS1 __attribute__((address_space(1)))
#define AS3 __attribute__((address_space(3)))

typedef int v4i __attribute__((vector_size(16)));   // 16-byte chunk type for b128 async copy

static constexpr int SEQ = 128, BATCH = 32, T = 32, T2 = 1024, T3 = 32768;
static constexpr int TILE_BYTES = T3 * 4;              // 128 KB per step tile
static constexpr size_t SMEM_BYTES = 2 * (size_t)TILE_BYTES + T2 * 4;  // 260 KB < 320 KB/WGP

__device__ __forceinline__ void async_copy_b128(const float* gsrc, uint32_t lds_off) {
#if __has_builtin(__builtin_amdgcn_global_load_async_to_lds_b128)
  __builtin_amdgcn_global_load_async_to_lds_b128(
      (AS1 v4i*)(uintptr_t)gsrc, (AS3 v4i*)lds_off, 0, 0);
#else
  asm volatile("global_load_async_to_lds_b128 %0, %1, off"
               :: "v"(lds_off), "v"((uint64_t)(uintptr_t)gsrc)
               : "memory");
#endif
}

__global__ __launch_bounds__(1024) void crf_scan_kernel(
    const float* __restrict__ scores,
    const unsigned char* __restrict__ mask,
    float* __restrict__ ws)
{
  extern __shared__ char smem[];
  float* tile0 = reinterpret_cast<float*>(smem);
  float* tile1 = reinterpret_cast<float*>(smem + TILE_BYTES);
  float* part  = reinterpret_cast<float*>(smem + 2 * TILE_BYTES);  // part[x*32+y], 4 KB

  const int b   = blockIdx.x;     // batch lane, 0..31
  const int tid = threadIdx.x;    // 0..1023
  const int w   = tid >> 5;       // wave index == output index y (0..31)
  const int l   = tid & 31;       // lane == output index z (0..31)

  // partition init: part[i,j] = scores[0,b,30,30,i] + scores[1,b,30,i,j]
  part[w * T + l] =
      scores[(size_t)(0 * BATCH + b) * T3 + 30 * T2 + 30 * T + w] +
      scores[(size_t)(1 * BATCH + b) * T3 + 30 * T2 + w * T + l];

  // prologue: kick off async copy of scores[2,b] into tile0 (8 x b128 per thread)
  {
    const float* g = scores + (size_t)(2 * BATCH + b) * T3;
    const uint32_t lbase = (uint32_t)(uintptr_t)tile0;
#pragma unroll
    for (int r = 0; r < 8; ++r) {
      const uint32_t c = (uint32_t)(r * 1024 + tid);           // 16B chunk index
      async_copy_b128(g + (size_t)c * 4, lbase + c * 16);
    }
  }
  __syncthreads();

  for (int t = 2; t < SEQ; ++t) {
    const float* cur = (t & 1) ? tile1 : tile0;

    // issue next tile's async copy, then wait for current tile (in-order ASYNCcnt)
    if (t + 1 < SEQ) {
      const float* g = scores + (size_t)((t + 1) * BATCH + b) * T3;
      float* nxt = ((t + 1) & 1) ? tile1 : tile0;
      const uint32_t lbase = (uint32_t)(uintptr_t)nxt;
#pragma unroll
      for (int r = 0; r < 8; ++r) {
        const uint32_t c = (uint32_t)(r * 1024 + tid);
        async_copy_b128(g + (size_t)c * 4, lbase + c * 16);
      }
      asm volatile("s_wait_asynccnt 0x8" ::: "memory");  // 8 outstanding = next tile only
    } else {
      asm volatile("s_wait_asynccnt 0x0" ::: "memory");  // drain for last tile
    }
    __syncthreads();  // whole tile visible to all waves

    // new_part[y=w, z=l] = logsumexp_x( tile[x,w,l] + part[x,w] )
    float v[T];
#pragma unroll
    for (int x = 0; x < T; ++x)
      v[x] = cur[x * T2 + w * T + l] + part[x * T + w];   // ds reads: conflict-free + broadcast
    float m = v[0];
#pragma unroll
    for (int x = 1; x < T; ++x) m = fmaxf(m, v[x]);
    float s = 0.0f;
#pragma unroll
    for (int x = 0; x < T; ++x) s += __expf(v[x] - m);
    const float lse = m + __logf(s);

    const bool mk = mask[t * BATCH + b] != 0;
    __syncthreads();                    // all reads of old partition done
    if (mk) part[w * T + l] = lse;
    __syncthreads();                    // new partition visible
  }

  if (tid == 0) ws[b] = part[31 * T + 31];   // partition[b, END, END]
}

__global__ __launch_bounds__(256) void crf_finalize_kernel(
    const float* __restrict__ scores,
    const long long* __restrict__ target,
    const unsigned char* __restrict__ mask,
    const float* __restrict__ ws,
    float* __restrict__ out)
{
  __shared__ float red[256];
  const int tid = threadIdx.x;

  // tg_energy: deterministic fixed-order partial sums + LDS tree reduction
  float acc = 0.0f;
  for (int k = tid; k < SEQ * BATCH; k += 256) {
    if (mask[k]) {
      const long long idx = target[k];
      acc += scores[(size_t)k * T3 + (size_t)idx];
    }
  }
  red[tid] = acc;
  __syncthreads();
#pragma unroll
  for (int s = 128; s > 0; s >>= 1) {
    if (tid < s) red[tid] += red[tid + s];
    __syncthreads();
  }

  if (tid == 0) {
    float z = 0.0f;
    for (int bb = 0; bb < BATCH; ++bb) z += ws[bb];      // fixed order: deterministic
    out[0] = (z - red[0]) / (float)BATCH;
  }
}

extern "C" void kernel_launch(void* const* d_in, const int* in_sizes, int n_in,
                              void* d_out, int out_size, void* d_ws, size_t ws_size,
                              hipStream_t stream) {
  (void)in_sizes; (void)n_in; (void)out_size; (void)ws_size;
  const float*         scores = (const float*)d_in[0];
  const long long*     target = (const long long*)d_in[1];
  const unsigned char* mask   = (const unsigned char*)d_in[2];
  float* out = (float*)d_out;
  float* ws  = (float*)d_ws;   // 32 floats of scratch

  // Opt in to >64KB dynamic LDS (gfx1250 supports 320KB/WGP). Host-side, non-stream,
  // deterministic, graph-capture-safe; ignore the result if the runtime doesn't need it.
  (void)hipFuncSetAttribute(reinterpret_cast<const void*>(&crf_scan_kernel),
                            hipFuncAttributeMaxDynamicSharedMemorySize,
                            (int)SMEM_BYTES);

  crf_scan_kernel<<<BATCH, 1024, SMEM_BYTES, stream>>>(scores, mask, ws);
  crf_finalize_kernel<<<1, 256, 0, stream>>>(scores, target, mask, ws, out);
}